// Comm_OUT_19378892439841
// MI455X (gfx1250) — compile-verified
//
#include <hip/hip_runtime.h>
#include <math.h>

// Problem constants: E=32,S=64 -> N=2048 rows; L=32; H=512; D_IN=1536; C=64.
#define NROWS 2048
#define DIN   1536
#define HID   512
#define LLEN  32
#define CDIM  64

typedef float v2f __attribute__((ext_vector_type(2)));
typedef float v8f __attribute__((ext_vector_type(8)));

#define WMMA_F32(A, B, C) \
  __builtin_amdgcn_wmma_f32_16x16x4_f32(false, (A), false, (B), (short)0, (C), false, false)

// ---------------------------------------------------------------------------
// V_WMMA_F32_16X16X4_F32 (wave32) fragment conventions:
//   A (16x4):  lane 0-15 -> M=lane, K={0,1}; lane 16-31 -> M=lane-16, K={2,3}
//   B (4x16):  lane 0-15 -> N=lane, K={0,1}; lane 16-31 -> N=lane-16, K={2,3}
//   C/D (16x16): VGPR r: lanes 0-15 -> M=r, N=lane; lanes 16-31 -> M=r+8
//
// Weight repack: B fragments need {B[k][n], B[k+1][n]} per lane. We repack
// every weight matrix so that pair is contiguous => one global_load_b64.
// ---------------------------------------------------------------------------

// ============ Repack K x N matrix into pair-interleaved layout =============
// dst[(k>>1)*(2N) + n*2 + (k&1)] = src[k*N + n]
__global__ void repack_pairs_kernel(const float* __restrict__ src,
                                    float* __restrict__ dst, int K, int N) {
  const int i = blockIdx.x * 256 + threadIdx.x;
  if (i >= K * N) return;
  const int k = i / N, n = i - k * N;
  dst[(size_t)(k >> 1) * (2 * N) + n * 2 + (k & 1)] = src[i];
}

// ====== Repack conv weights: C'[slab(g,j)][co][ci] = w_g[co][ci][j] ========
// Slab = 128*512 floats; slab base per group: g0->0, g1->1, g2->4, g3->9.
__global__ void repack_conv_kernel(const float* __restrict__ w1,
                                   const float* __restrict__ w3,
                                   const float* __restrict__ w5,
                                   const float* __restrict__ w7,
                                   float* __restrict__ dst) {
  const int i = blockIdx.x * 256 + threadIdx.x;   // 16 * 65536 elements
  const int slab = i >> 16;
  const int within = i & 65535;
  const int co = within >> 9, ci = within & 511;
  int g, j;
  if (slab < 1)      { g = 0; j = slab; }
  else if (slab < 4) { g = 1; j = slab - 1; }
  else if (slab < 9) { g = 2; j = slab - 4; }
  else               { g = 3; j = slab - 9; }
  const int ksz = 2 * g + 1;
  const float* w = (g == 0) ? w1 : (g == 1) ? w3 : (g == 2) ? w5 : w7;
  dst[i] = w[((size_t)co * HID + ci) * ksz + j];
}

// ======================= Kernel 1: xw = x @ Wx + bx ========================
// M=2048, K=1536, N=512.  One wave per 32x32 tile (2x2 register blocking).
// Wxp is pair-interleaved: fragment = one b64 per B sub-tile.
__global__ void gemm_xw_kernel(const float* __restrict__ X,
                               const float* __restrict__ Wxp,
                               const float* __restrict__ bx,
                               float* __restrict__ xw) {
  const int tile  = blockIdx.x * 8 + (threadIdx.x >> 5);  // 1024 tiles
  const int mtile = tile >> 4;                            // 64 M-tiles of 32
  const int ntile = tile & 15;                            // 16 N-tiles of 32
  const int lane  = threadIdx.x & 31;
  const int half  = lane >> 4, mr = lane & 15, k0 = half * 2;
  const int m0 = mtile * 32, n0 = ntile * 32;

  const float* arow0 = X + (size_t)(m0 + mr) * DIN;
  const float* arow1 = X + (size_t)(m0 + 16 + mr) * DIN;
  // pair-layout base for this lane's column: &Wxp[(k0>>1)*1024 + (n0+mr)*2]
  const float* bcol = Wxp + (size_t)half * (2 * HID) + (size_t)(n0 + mr) * 2;
  v8f acc00 = {}, acc01 = {}, acc10 = {}, acc11 = {};
  for (int kk = 0; kk < DIN; kk += 4) {
    v2f a0 = {arow0[kk + k0], arow0[kk + k0 + 1]};
    v2f a1 = {arow1[kk + k0], arow1[kk + k0 + 1]};
    const float* bp = bcol + (size_t)(kk >> 1) * (2 * HID);
    v2f b0 = {bp[0],  bp[1]};      // one b64
    v2f b1 = {bp[32], bp[33]};     // one b64 (16 columns ahead)
    acc00 = WMMA_F32(a0, b0, acc00);
    acc01 = WMMA_F32(a0, b1, acc01);
    acc10 = WMMA_F32(a1, b0, acc10);
    acc11 = WMMA_F32(a1, b1, acc11);
  }
  for (int r = 0; r < 8; ++r) {
    const int row = m0 + r + half * 8;
    const int col = n0 + mr;
    xw[(size_t)row * HID + col]             = acc00[r] + bx[col];
    xw[(size_t)row * HID + col + 16]        = acc01[r] + bx[col + 16];
    xw[(size_t)(row + 16) * HID + col]      = acc10[r] + bx[col];
    xw[(size_t)(row + 16) * HID + col + 16] = acc11[r] + bx[col + 16];
  }
}

// ======================= Kernel 2: h0 = tanh(xw + bh) ======================
__global__ void h0_kernel(const float* __restrict__ xw,
                          const float* __restrict__ bh,
                          float* __restrict__ h0) {
  const int i = blockIdx.x * 256 + threadIdx.x;           // NROWS*HID threads
  h0[i] = tanhf(xw[i] + bh[i & (HID - 1)]);
}

// ============ Kernel 3: h_l = tanh(xw + h_{l-1} @ Wh + bh) =================
// M=2048, K=512, N=512; 32x32 tile per wave; Whp pair-interleaved.
__global__ void rnn_step_kernel(const float* __restrict__ hprev,
                                const float* __restrict__ Whp,
                                const float* __restrict__ bh,
                                const float* __restrict__ xw,
                                float* __restrict__ hout) {
  const int tile  = blockIdx.x * 8 + (threadIdx.x >> 5);  // 1024 tiles
  const int mtile = tile >> 4;
  const int ntile = tile & 15;
  const int lane  = threadIdx.x & 31;
  const int half  = lane >> 4, mr = lane & 15, k0 = half * 2;
  const int m0 = mtile * 32, n0 = ntile * 32;

  const float* arow0 = hprev + (size_t)(m0 + mr) * HID;
  const float* arow1 = hprev + (size_t)(m0 + 16 + mr) * HID;
  const float* bcol = Whp + (size_t)half * (2 * HID) + (size_t)(n0 + mr) * 2;
  v8f acc00 = {}, acc01 = {}, acc10 = {}, acc11 = {};
  for (int kk = 0; kk < HID; kk += 4) {
    v2f a0 = {arow0[kk + k0], arow0[kk + k0 + 1]};
    v2f a1 = {arow1[kk + k0], arow1[kk + k0 + 1]};
    const float* bp = bcol + (size_t)(kk >> 1) * (2 * HID);
    v2f b0 = {bp[0],  bp[1]};
    v2f b1 = {bp[32], bp[33]};
    acc00 = WMMA_F32(a0, b0, acc00);
    acc01 = WMMA_F32(a0, b1, acc01);
    acc10 = WMMA_F32(a1, b0, acc10);
    acc11 = WMMA_F32(a1, b1, acc11);
  }
  for (int r = 0; r < 8; ++r) {
    const int row = m0 + r + half * 8;
    const int col = n0 + mr;
    const size_t i00 = (size_t)row * HID + col;
    const size_t i10 = (size_t)(row + 16) * HID + col;
    hout[i00]      = tanhf(acc00[r] + xw[i00]      + bh[col]);
    hout[i00 + 16] = tanhf(acc01[r] + xw[i00 + 16] + bh[col + 16]);
    hout[i10]      = tanhf(acc10[r] + xw[i10]      + bh[col]);
    hout[i10 + 16] = tanhf(acc11[r] + xw[i10 + 16] + bh[col + 16]);
  }
}

// ===== Kernel 4: fused Ted_Conv1d (k=1,3,5,7, reflect pad) + PReLU + Wout ==
// Grid: (128 row-tiles of 16, 32 positions l). Block = 256 threads (8 waves).
// Phase 1: wave w owns 4 consecutive cout-tiles inside conv group g = w>>1;
//          one A fragment (from hs[reflect(l+j-p)]) per K-step feeds 4 wmmas;
//          B fragments are contiguous b64 from the repacked conv weights.
// Phase 2: waves 0-3 compute pred[16 x 64] = Y @ Wout + bout (Wout repacked).
__global__ void conv_out_kernel(const float* __restrict__ hs,
                                const float* __restrict__ cpack,
                                const float* __restrict__ b1,
                                const float* __restrict__ b3,
                                const float* __restrict__ b5,
                                const float* __restrict__ b7,
                                const float* __restrict__ prelu_a,
                                const float* __restrict__ Woutp,
                                const float* __restrict__ bout,
                                float* __restrict__ out) {
  __shared__ float Yt[16][HID + 8];    // padded to dodge LDS bank conflicts

  const int l = blockIdx.y;
  const int nbase = blockIdx.x * 16;
  const int wave = threadIdx.x >> 5;
  const int lane = threadIdx.x & 31;
  const int half = lane >> 4, mr = lane & 15, k0 = half * 2;
  const float alpha = prelu_a[0];

  const float* cb[4] = {b1, b3, b5, b7};

  // Wave -> (group, 4 tiles): g = wave>>1; tiles sub..sub+3; sub=(wave&1)*4.
  const int g        = wave >> 1;
  const int ksz      = 2 * g + 1;            // kernel size 1/3/5/7, pad = g
  const int sub      = (wave & 1) * 4;
  const int colocal0 = sub * 16;
  const int slab0    = g * g;                // 0,1,4,9: first slab of group
  // Repacked: cpack[slab][co_local][ci], slab = slab0 + j
  const float* Bg = cpack + (size_t)slab0 * (128 * HID);

  v8f acc[4] = {{}, {}, {}, {}};
  for (int j = 0; j < ksz; ++j) {
    int lr = l + j - g;                      // reflect padding
    if (lr < 0) lr = -lr;
    if (lr >= LLEN) lr = 2 * LLEN - 2 - lr;
    const float* arow =
        hs + (size_t)lr * NROWS * HID + (size_t)(nbase + mr) * HID;
    const float* Brow0 =
        Bg + (size_t)j * (128 * HID) + (size_t)(colocal0 + mr) * HID;
    for (int kk = 0; kk < HID; kk += 4) {
      v2f a = {arow[kk + k0], arow[kk + k0 + 1]};
      for (int t = 0; t < 4; ++t) {
        const float* Bp = Brow0 + (size_t)t * (16 * HID) + kk + k0;
        v2f b = {Bp[0], Bp[1]};              // one b64
        acc[t] = WMMA_F32(a, b, acc[t]);
      }
    }
  }
  for (int t = 0; t < 4; ++t) {
    const int co0 = (g << 7) + (sub + t) * 16;       // global cout base
    const float bias = cb[g][colocal0 + t * 16 + mr];
    for (int r = 0; r < 8; ++r) {
      float v = acc[t][r] + bias;
      v = (v >= 0.0f) ? v : alpha * v;               // PReLU
      Yt[r + half * 8][co0 + mr] = v;
    }
  }
  __syncthreads();

  if (wave < 4) {                                    // 4 tiles cover C=64
    const int n0 = wave * 16;
    const float* bcol =
        Woutp + (size_t)half * (2 * CDIM) + (size_t)(n0 + mr) * 2;
    v8f a2 = {};
    for (int kk = 0; kk < HID; kk += 4) {
      v2f a = {Yt[mr][kk + k0], Yt[mr][kk + k0 + 1]};
      const float* bp = bcol + (size_t)(kk >> 1) * (2 * CDIM);
      v2f b = {bp[0], bp[1]};                        // one b64
      a2 = WMMA_F32(a, b, a2);
    }
    for (int r = 0; r < 8; ++r) {
      const int row = nbase + r + half * 8;          // sample index n
      const int c = n0 + mr;
      out[((size_t)row * LLEN + l) * CDIM + c] = a2[r] + bout[c];
    }
  }
}

// ========== Kernel 5: end-token mask (argmax==0, keep through first) =======
__global__ void mask_kernel(float* __restrict__ out) {
  const int n = blockIdx.x * 8 + (threadIdx.x >> 5);      // 2048 waves
  const int l = threadIdx.x & 31;
  float* row = out + ((size_t)n * LLEN + l) * CDIM;
  const float v0 = row[0];
  float mx = row[1];
  for (int c = 2; c < CDIM; ++c) mx = fmaxf(mx, row[c]);
  const bool is_end = (v0 >= mx);               // argmax ties -> index 0
  const unsigned long long bal = __ballot(is_end ? 1 : 0);
  const bool keep = ((bal & ((1ull << l) - 1ull)) == 0ull);
  if (!keep) {
    for (int c = 0; c < CDIM; ++c) row[c] = 0.0f;
  }
}

// ===========================================================================
extern "C" void kernel_launch(void* const* d_in, const int* in_sizes, int n_in,
                              void* d_out, int out_size, void* d_ws, size_t ws_size,
                              hipStream_t stream) {
  (void)in_sizes; (void)n_in; (void)out_size; (void)ws_size;

  const float* hwa  = (const float*)d_in[0];   // (2048,1536)
  const float* Wx   = (const float*)d_in[1];   // (1536,512)
  const float* bx   = (const float*)d_in[2];
  const float* Wh   = (const float*)d_in[3];   // (512,512)
  const float* bh   = (const float*)d_in[4];
  const float* pa   = (const float*)d_in[5];
  const float* Wout = (const float*)d_in[6];   // (512,64)
  const float* bout = (const float*)d_in[7];
  const float* w1 = (const float*)d_in[8],  *b1 = (const float*)d_in[9];
  const float* w3 = (const float*)d_in[10], *b3 = (const float*)d_in[11];
  const float* w5 = (const float*)d_in[12], *b5 = (const float*)d_in[13];
  const float* w7 = (const float*)d_in[14], *b7 = (const float*)d_in[15];

  float* out = (float*)d_out;                  // (2048, 32, 64)
  float* ws  = (float*)d_ws;

  // Workspace layout (floats):
  float* xw    = ws;                                   // 2048*512
  float* hs    = xw   + (size_t)NROWS * HID;           // 32*2048*512
  float* Wxp   = hs   + (size_t)LLEN * NROWS * HID;    // 1536*512
  float* Whp   = Wxp  + (size_t)DIN * HID;             // 512*512
  float* Woutp = Whp  + (size_t)HID * HID;             // 512*64
  float* Cpack = Woutp + (size_t)HID * CDIM;           // 16*128*512

  // 0) one-time weight repacks into WMMA-fragment-native layouts
  repack_pairs_kernel<<<(DIN * HID) / 256, 256, 0, stream>>>(Wx, Wxp, DIN, HID);
  repack_pairs_kernel<<<(HID * HID) / 256, 256, 0, stream>>>(Wh, Whp, HID, HID);
  repack_pairs_kernel<<<(HID * CDIM) / 256, 256, 0, stream>>>(Wout, Woutp, HID, CDIM);
  repack_conv_kernel<<<(16 * 128 * HID) / 256, 256, 0, stream>>>(w1, w3, w5, w7, Cpack);

  // 1) xw = x @ Wx + bx   (1024 wave-tiles of 32x32, 8 waves/block)
  gemm_xw_kernel<<<128, 256, 0, stream>>>(hwa, Wxp, bx, xw);

  // 2) hs[0] = tanh(xw + bh)
  h0_kernel<<<(NROWS * HID) / 256, 256, 0, stream>>>(xw, bh, hs);

  // 3) hs[l] = tanh(xw + hs[l-1] @ Wh + bh), l = 1..31 (sequential)
  for (int l = 1; l < LLEN; ++l) {
    rnn_step_kernel<<<128, 256, 0, stream>>>(
        hs + (size_t)(l - 1) * NROWS * HID, Whp, bh, xw,
        hs + (size_t)l * NROWS * HID);
  }

  // 4) fused conv(1,3,5,7 reflect) + PReLU + output projection
  conv_out_kernel<<<dim3(NROWS / 16, LLEN), 256, 0, stream>>>(
      hs, Cpack, b1, b3, b5, b7, pa, Woutp, bout, out);

  // 5) end-token mask
  mask_kernel<<<NROWS / 8, 256, 0, stream>>>(out);
}